// SelfAttention_11587821764716
// MI455X (gfx1250) — compile-verified
//
#include <hip/hip_runtime.h>

typedef __attribute__((ext_vector_type(2))) float v2f;
typedef __attribute__((ext_vector_type(8))) float v8f;

// D = A(16x4,f32) * B(4x16,f32) + C(16x16,f32)   -> v_wmma_f32_16x16x4_f32
__device__ __forceinline__ v8f wmma_f32_k4(v2f a, v2f b, v8f c) {
  return __builtin_amdgcn_wmma_f32_16x16x4_f32(
      /*neg_a=*/false, a, /*neg_b=*/false, b,
      /*c_mod=*/(short)0, c, /*reuse_a=*/false, /*reuse_b=*/false);
}

// ---------------------------------------------------------------------------
// NT GEMM: C[M,N] = A[M,K] * B[N,K]^T   (both row-major, contiguous K)
// Grid: (M/64, N/64); block = 128 threads = 4 waves; each wave does 32x32.
// Fragment mapping (f32 WMMA):
//   A 16x4 : lane holds (m = lane%16, k = 2*(lane/16)+{0,1})  -> one float2
//   B 4x16 : lane holds (n = lane%16, k = 2*(lane/16)+{0,1})  -> one float2
//   C 16x16: VGPR r holds (m = r + 8*(lane/16), n = lane%16)
// ---------------------------------------------------------------------------
__global__ __launch_bounds__(128) void gemm_nt_f32_wmma(
    const float* __restrict__ A, int lda,
    const float* __restrict__ B, int ldb,
    float* __restrict__ C, int ldc, int K) {
  const int wave = threadIdx.x >> 5;
  const int lane = threadIdx.x & 31;
  const int lm = lane & 15;
  const int lk = lane >> 4;
  const int m0 = blockIdx.x * 64 + (wave & 1) * 32;
  const int n0 = blockIdx.y * 64 + (wave >> 1) * 32;

  const float* a0p = A + (size_t)(m0 + lm) * lda + 2 * lk;
  const float* a1p = a0p + (size_t)16 * lda;
  const float* b0p = B + (size_t)(n0 + lm) * ldb + 2 * lk;
  const float* b1p = b0p + (size_t)16 * ldb;

  v8f c00 = {}, c01 = {}, c10 = {}, c11 = {};
#pragma unroll 4
  for (int k = 0; k < K; k += 4) {
    v2f a0 = *(const v2f*)(a0p + k);
    v2f a1 = *(const v2f*)(a1p + k);
    v2f b0 = *(const v2f*)(b0p + k);
    v2f b1 = *(const v2f*)(b1p + k);
    c00 = wmma_f32_k4(a0, b0, c00);
    c01 = wmma_f32_k4(a0, b1, c01);
    c10 = wmma_f32_k4(a1, b0, c10);
    c11 = wmma_f32_k4(a1, b1, c11);
  }

  const int row = 8 * lk;
#pragma unroll
  for (int r = 0; r < 8; ++r) {
    C[(size_t)(m0 + row + r) * ldc + (n0 + lm)]           = c00[r];
    C[(size_t)(m0 + row + r) * ldc + (n0 + 16 + lm)]      = c01[r];
    C[(size_t)(m0 + 16 + row + r) * ldc + (n0 + lm)]      = c10[r];
    C[(size_t)(m0 + 16 + row + r) * ldc + (n0 + 16 + lm)] = c11[r];
  }
}

// ---------------------------------------------------------------------------
// Flash attention, fp32 WMMA. qkv: [B*S, 3*H*D] (f = [3][H][D]); ctx: [B*S, H*D]
// One wave handles one (b, h, 16-row q-tile); streams keys in blocks of 16.
// ---------------------------------------------------------------------------
__global__ __launch_bounds__(256) void flash_attn_f32_wmma(
    const float* __restrict__ qkv, float* __restrict__ ctx) {
  constexpr int S = 2048, H = 16, Dh = 64;
  constexpr int QKV_LD = 3 * H * Dh;  // 3072
  constexpr int CTX_LD = H * Dh;      // 1024
  constexpr float SCALE = 0.125f;     // 64^{-1/2}

  __shared__ float p_lds[8][16 * 16];  // per-wave P-tile bounce buffer (8 KB)

  const int wave = threadIdx.x >> 5;
  const int lane = threadIdx.x & 31;
  const int lm = lane & 15;
  const int lk = lane >> 4;

  const int gw = blockIdx.x * 8 + wave;  // [0, B*H*S/16) = [0, 8192)
  const int qt = gw & 127;               // q tile within sequence
  const int bh = gw >> 7;                // b*H + h
  const int h  = bh & 15;
  const int b  = bh >> 4;

  const float* qbase = qkv + (size_t)b * S * QKV_LD + h * Dh;
  const float* kbase = qbase + H * Dh;       // +1024
  const float* vbase = qbase + 2 * H * Dh;   // +2048

  // Q fragments (A-layout), 16 chunks of K=4 covering Dh=64 — resident in VGPRs
  v2f qa[16];
  {
    const float* qrow = qbase + (size_t)(qt * 16 + lm) * QKV_LD + 2 * lk;
#pragma unroll
    for (int kc = 0; kc < 16; ++kc) qa[kc] = *(const v2f*)(qrow + 4 * kc);
  }

  v8f acc[4] = {};                 // ctx accumulator, C-layout, 16 x 64
  float mrow[8], lrow[8];          // per-lane row stats: row = 8*lk + r
#pragma unroll
  for (int r = 0; r < 8; ++r) { mrow[r] = -__builtin_inff(); lrow[r] = 0.0f; }

  float* myp = p_lds[wave];

  for (int j = 0; j < S; j += 16) {
    // ---- scores tile S[16q x 16k] = Q * K^T -------------------------------
    v8f s = {};
    const float* krow = kbase + (size_t)(j + lm) * QKV_LD + 2 * lk;
#pragma unroll
    for (int kc = 0; kc < 16; ++kc) {
      v2f bf = *(const v2f*)(krow + 4 * kc);
      s = wmma_f32_k4(qa[kc], bf, s);
    }

    // ---- online softmax (row reductions across the 16-lane half) ---------
    float corr[8];
#pragma unroll
    for (int r = 0; r < 8; ++r) {
      float sv = s[r] * SCALE;
      float mx = sv;
#pragma unroll
      for (int off = 8; off >= 1; off >>= 1)
        mx = fmaxf(mx, __shfl_xor(mx, off, 16));
      const float mnew = fmaxf(mrow[r], mx);
      const float p = __expf(sv - mnew);
      corr[r] = __expf(mrow[r] - mnew);
      float ps = p;
#pragma unroll
      for (int off = 8; off >= 1; off >>= 1)
        ps += __shfl_xor(ps, off, 16);
      lrow[r] = lrow[r] * corr[r] + ps;
      mrow[r] = mnew;
      s[r] = p;
    }
#pragma unroll
    for (int t = 0; t < 4; ++t)
#pragma unroll
      for (int r = 0; r < 8; ++r) acc[t][r] *= corr[r];

    // ---- transpose P: C-layout -> A-layout via LDS (same-wave, no barrier)
#pragma unroll
    for (int r = 0; r < 8; ++r) myp[(8 * lk + r) * 16 + lm] = s[r];
    v2f pa[4];
#pragma unroll
    for (int kc = 0; kc < 4; ++kc)
      pa[kc] = *(const v2f*)(myp + lm * 16 + 4 * kc + 2 * lk);

    // ---- ctx += P(16x16) * V(16x64) ---------------------------------------
#pragma unroll
    for (int kc = 0; kc < 4; ++kc) {
      const float* vrow = vbase + (size_t)(j + 4 * kc + 2 * lk) * QKV_LD + lm;
#pragma unroll
      for (int t = 0; t < 4; ++t) {
        v2f bf;
        bf.x = vrow[t * 16];           // k = 4*kc + 2*lk,     n = t*16 + lm
        bf.y = vrow[QKV_LD + t * 16];  // k = 4*kc + 2*lk + 1
        acc[t] = wmma_f32_k4(pa[kc], bf, acc[t]);
      }
    }
  }

  // ---- normalize and store ctx in [b, s, h*Dh] layout ----------------------
  float* crow = ctx + (size_t)(b * S + qt * 16 + 8 * lk) * CTX_LD + h * Dh + lm;
#pragma unroll
  for (int t = 0; t < 4; ++t) {
#pragma unroll
    for (int r = 0; r < 8; ++r) {
      crow[(size_t)r * CTX_LD + t * 16] = acc[t][r] / lrow[r];
    }
  }
}

// ---------------------------------------------------------------------------
extern "C" void kernel_launch(void* const* d_in, const int* in_sizes, int n_in,
                              void* d_out, int out_size, void* d_ws, size_t ws_size,
                              hipStream_t stream) {
  (void)in_sizes; (void)n_in; (void)out_size; (void)ws_size;

  const float* x     = (const float*)d_in[0];  // [4, 2048, 1024]
  const float* W_qkv = (const float*)d_in[1];  // [3072, 1024]
  const float* W_out = (const float*)d_in[2];  // [1024, 1024]
  float* out = (float*)d_out;                  // [4, 2048, 1024]

  constexpr int M    = 4 * 2048;   // 8192 rows
  constexpr int HID  = 1024;
  constexpr int NQKV = 3 * HID;    // 3072

  float* qkv_ws = (float*)d_ws;                    // 8192*3072*4 = 96 MB
  float* ctx_ws = qkv_ws + (size_t)M * NQKV;       // 8192*1024*4 = 32 MB

  // 1) fused QKV projection: qkv = x * W_qkv^T
  gemm_nt_f32_wmma<<<dim3(M / 64, NQKV / 64), 128, 0, stream>>>(
      x, HID, W_qkv, HID, qkv_ws, NQKV, HID);

  // 2) flash attention: ctx[b,s,h*d]
  flash_attn_f32_wmma<<<dim3((4 * 16 * (2048 / 16)) / 8), 256, 0, stream>>>(
      qkv_ws, ctx_ws);

  // 3) output projection: out = ctx * W_out^T  (K = HID)
  gemm_nt_f32_wmma<<<dim3(M / 64, HID / 64), 128, 0, stream>>>(
      ctx_ws, HID, W_out, HID, out, HID, HID);
}